// PlanesweepCorrelation_67370857005293
// MI455X (gfx1250) — compile-verified
//
#include <hip/hip_runtime.h>
#include <hip/hip_bf16.h>
#include <math.h>

// Problem constants (from reference): N,C,S,H,W = 2,64,32,128,160; G=8, gc=8
#define Nn 2
#define Cc 64
#define Ss 32
#define Hh 128
#define Ww 160
#define Gg 8
#define HWc (Hh * Ww)

typedef __attribute__((ext_vector_type(2))) float v2f;
typedef __attribute__((ext_vector_type(8))) float v8f;

// ---------------------------------------------------------------------------
// Pass 1: channel-wise L2 normalization of feat_ref / feat_src into workspace.
// One thread per spatial position; both tensors handled in one grid.
// Features (2 x 10.5 MB) then live in L2 (192 MB) for the gather-heavy pass 2.
// ---------------------------------------------------------------------------
__global__ void __launch_bounds__(256) psweep_normalize(
    const float* __restrict__ fref, const float* __restrict__ fsrc,
    float* __restrict__ oref, float* __restrict__ osrc)
{
    int idx = blockIdx.x * blockDim.x + threadIdx.x;   // [0, 2*N*HW)
    if (idx >= 2 * Nn * HWc) return;
    int which = idx / (Nn * HWc);
    int p     = idx % (Nn * HWc);
    const float* in  = which ? fsrc : fref;
    float*       out = which ? osrc : oref;
    const float* ip = in  + (size_t)(p / HWc) * Cc * HWc + (p % HWc);
    float*       op = out + (size_t)(p / HWc) * Cc * HWc + (p % HWc);

    float ss = 0.f;
#pragma unroll
    for (int c = 0; c < Cc; ++c) {
        float v = ip[c * HWc];
        ss += v * v;
    }
    float inv = 1.f / (sqrtf(ss) + 1e-9f);
#pragma unroll
    for (int c = 0; c < Cc; ++c)
        op[c * HWc] = ip[c * HWc] * inv;
}

// ---------------------------------------------------------------------------
// Pass 2: one wave per pixel (n,h,w). Per pixel:
//   corr[s,g] = sum_c W[s,c] * B[c,g],  B[c,g] = ref_n[c] * (g == c/8)
// computed as 2 M-tiles x 16 K-steps of V_WMMA_F32_16X16X4_F32.
// Lane l supplies, for K-step k and register r, channel c = 4k + r + 2*(l/16):
//   A reg r : bilinear warp of src_n at its own sample s = l%16 + 16t
//   B reg r : ref_n[c] masked to lane group g == k/2
// D layout: lane l holds (g = l%16, s = r + 8*(l/16) + 16t).
// ---------------------------------------------------------------------------
__global__ void __launch_bounds__(256) psweep_wmma(
    const float* __restrict__ grids,
    const float* __restrict__ refn,
    const float* __restrict__ srcn,
    float* __restrict__ corr,
    float* __restrict__ wmask)
{
    const int lane = threadIdx.x & 31;
    const int wave = blockIdx.x * (blockDim.x >> 5) + (threadIdx.x >> 5);
    const int n  = wave / HWc;
    const int hw = wave % HWc;
    const int hi = lane >> 4;     // lane half: 0 or 1
    const int lg = lane & 15;     // lane within half

    const float* refp = refn + (size_t)n * Cc * HWc + hw;   // refp[c*HW]
    const float* srcp = srcn + (size_t)n * Cc * HWc;

#pragma unroll
    for (int t = 0; t < 2; ++t) {
        const int s = lg + 16 * t;   // this lane's sample for the A matrix
        // grids[n, s, axis, h, w] — streamed once, keep out of caches (NT)
        const float gx = __builtin_nontemporal_load(
            grids + (((size_t)n * Ss + s) * 2 + 0) * HWc + hw);
        const float gy = __builtin_nontemporal_load(
            grids + (((size_t)n * Ss + s) * 2 + 1) * HWc + hw);
        const float ix = gx - 0.5f, iy = gy - 0.5f;
        const float fx0 = floorf(ix), fy0 = floorf(iy);
        const int x0 = (int)fx0, y0 = (int)fy0;
        const float wx1 = ix - fx0, wy1 = iy - fy0;
        const float wx0 = 1.f - wx1, wy0 = 1.f - wy1;

        const float v00 = (x0   >= 0 && x0   < Ww && y0   >= 0 && y0   < Hh) ? 1.f : 0.f;
        const float v10 = (x0+1 >= 0 && x0+1 < Ww && y0   >= 0 && y0   < Hh) ? 1.f : 0.f;
        const float v01 = (x0   >= 0 && x0   < Ww && y0+1 >= 0 && y0+1 < Hh) ? 1.f : 0.f;
        const float v11 = (x0+1 >= 0 && x0+1 < Ww && y0+1 >= 0 && y0+1 < Hh) ? 1.f : 0.f;

        const float w00 = wx0 * wy0 * v00, w10 = wx1 * wy0 * v10;
        const float w01 = wx0 * wy1 * v01, w11 = wx1 * wy1 * v11;

        const int xc0 = min(max(x0,     0), Ww - 1);
        const int xc1 = min(max(x0 + 1, 0), Ww - 1);
        const int yc0 = min(max(y0,     0), Hh - 1);
        const int yc1 = min(max(y0 + 1, 0), Hh - 1);
        const int o00 = yc0 * Ww + xc0, o10 = yc0 * Ww + xc1;
        const int o01 = yc1 * Ww + xc0, o11 = yc1 * Ww + xc1;

        // warp-validity mask (grid_sample of ones, thresholded)
        const float msum = w00 + w10 + w01 + w11;
        const float wm = (msum < 0.9999f) ? 0.f : 1.f;

        v8f acc = {};
#pragma unroll
        for (int k = 0; k < 16; ++k) {
            const int c0 = 4 * k + 2 * hi;         // channel for register 0
            const float* p0 = srcp + (size_t)c0 * HWc;
            const float* p1 = p0 + HWc;
            v2f a, b;
            a.x = w00 * p0[o00] + w10 * p0[o10] + w01 * p0[o01] + w11 * p0[o11];
            a.y = w00 * p1[o00] + w10 * p1[o10] + w01 * p1[o01] + w11 * p1[o11];
            const float bm = (lg == (k >> 1)) ? 1.f : 0.f;   // block-diagonal B
            b.x = bm * refp[(size_t)c0 * HWc];
            b.y = bm * refp[(size_t)(c0 + 1) * HWc];
            acc = __builtin_amdgcn_wmma_f32_16x16x4_f32(
                false, a, false, b, (short)0, acc, false, false);
        }

        // Apply per-sample mask and store. D element r of lane l is
        // (g = lg, s = r + 8*hi + 16t); mask for that s lives in lane r+8*hi.
#pragma unroll
        for (int r = 0; r < 8; ++r) {
            const float mr = __shfl(wm, r + 8 * hi, 32);   // convergent
            if (lg < Gg) {
                const int sg = r + 8 * hi + 16 * t;
                __builtin_nontemporal_store(
                    acc[r] * mr,
                    corr + (((size_t)n * Gg + lg) * Ss + sg) * HWc + hw);
            }
        }
        if (hi == 0) {
            __builtin_nontemporal_store(
                wm, wmask + ((size_t)n * Ss + s) * HWc + hw);
        }
    }
}

extern "C" void kernel_launch(void* const* d_in, const int* in_sizes, int n_in,
                              void* d_out, int out_size, void* d_ws, size_t ws_size,
                              hipStream_t stream) {
    const float* feat_ref = (const float*)d_in[0];   // [N,C,H,W]
    const float* feat_src = (const float*)d_in[1];   // [N,C,H,W]
    const float* grids    = (const float*)d_in[2];   // [N,S,2,H,W]
    // d_in[3] = num_groups (==8, baked in at compile time)

    float* corr  = (float*)d_out;                               // [N,G,S,H,W]
    float* wmask = corr + (size_t)Nn * Gg * Ss * HWc;           // [N,S,H,W]

    float* refn = (float*)d_ws;                                 // [N,C,H,W]
    float* srcn = refn + (size_t)Nn * Cc * HWc;                 // [N,C,H,W]

    const int norm_threads = 2 * Nn * HWc;                      // 81,920
    psweep_normalize<<<(norm_threads + 255) / 256, 256, 0, stream>>>(
        feat_ref, feat_src, refn, srcn);

    const int waves = Nn * HWc;                                 // 40,960 pixels
    psweep_wmma<<<waves / 8, 256, 0, stream>>>(grids, refn, srcn, corr, wmask);
}